// MAR_missingness_83992380440895
// MI455X (gfx1250) — compile-verified
//
#include <hip/hip_runtime.h>
#include <math.h>

typedef __attribute__((ext_vector_type(2))) float v2f;
typedef __attribute__((ext_vector_type(8))) float v8f;

#define NN   256
#define TT   64
#define HM   32
#define KIN  81            // 9*9 flattened patch
#define WPB  8             // waves per block (wave32)

__global__ __launch_bounds__(256) void mar_mlp_kernel(
    const float* __restrict__ X,
    const int*   __restrict__ rows_idx,
    const int*   __restrict__ cols_idx,
    const float* __restrict__ W_in,
    const float* __restrict__ b_in,
    const float* __restrict__ W_hid,
    const float* __restrict__ b_hid,
    const float* __restrict__ W_out,
    const float* __restrict__ b_out,
    float*       __restrict__ out)
{
    __shared__ float lds[WPB * 192];   // per wave: V[84] h0[32] h1[32] h2[32]

    const int lane = threadIdx.x & 31;
    const int wave = threadIdx.x >> 5;
    const int cell = blockIdx.x * WPB + wave;     // 0 .. N*T-1
    const int i    = cell / TT;
    const int t    = cell - i * TT;

    float* V  = &lds[wave * 192];
    float* h0 = V + 84;
    float* h1 = h0 + 32;
    float* h2 = h1 + 32;

    // ---- gather V[r*9+c] = X[rows[r], cols[c]] ----
    const int* rp = rows_idx + cell * 9;
    const int* cp = cols_idx + cell * 9;
    for (int idx = lane; idx < KIN; idx += 32) {
        int r = idx / 9, c = idx - r * 9;
        V[idx] = X[rp[r] * TT + cp[c]];
    }
    __builtin_amdgcn_wave_barrier();

    const int hf   = lane >> 4;        // half-wave id: 0 (lanes 0-15), 1 (lanes 16-31)
    const int mrow = lane & 15;        // row within 16x16 tile
    const int koff = hf * 2;           // K offset carried by upper half-wave

    const float* Wi = W_in + (size_t)cell * (HM * KIN);
    const float* bi = b_in + (size_t)cell * HM;

    // ---- layer 0: h = relu(W_in[32x81] @ V + b_in) via V_WMMA_F32_16X16X4_F32 ----
    float hreg[2][8];
    const float v80 = V[80];
    for (int ho = 0; ho < 2; ++ho) {
        const int halfbase = ho * 16;
        v8f c;
        #pragma unroll
        for (int j = 0; j < 8; ++j) {
            int m = halfbase + hf * 8 + j;           // C layout: VGPRj -> M=j (lo) / 8+j (hi)
            c[j] = bi[m] + Wi[m * KIN + 80] * v80;   // fold K=80 tail into accumulator init
        }
        const float* Arow = Wi + (halfbase + mrow) * KIN + koff;
        #pragma unroll
        for (int kc = 0; kc < 20; ++kc) {            // K = 0..79 in chunks of 4
            v2f a; a.x = Arow[kc * 4];           a.y = Arow[kc * 4 + 1];
            v2f b; b.x = V[kc * 4 + koff];       b.y = V[kc * 4 + koff + 1];
            c = __builtin_amdgcn_wmma_f32_16x16x4_f32(false, a, false, b,
                                                      (short)0, c, false, false);
        }
        #pragma unroll
        for (int j = 0; j < 8; ++j) {
            float v = c[j];
            hreg[ho][j] = v > 0.0f ? v : 0.0f;       // relu
        }
    }
    // column 0 of each half holds the matvec result -> spill h (32 floats) to LDS
    if (mrow == 0) {
        #pragma unroll
        for (int j = 0; j < 8; ++j) {
            h0[hf * 8 + j]      = hreg[0][j];
            h0[16 + hf * 8 + j] = hreg[1][j];
        }
    }
    __builtin_amdgcn_wave_barrier();   // same-wave DS ops are in-order; just fence the compiler

    // ---- hidden layers k=0,1: h = W_hid[k][32x32] @ h + b_hid[k] (relu after k=0) ----
    const float* hin = h0;
    float* hout = h1;
    #pragma unroll
    for (int layer = 0; layer < 2; ++layer) {
        const float* Wl = W_hid + (((size_t)layer * NN + i) * TT + t) * (HM * HM);
        const float* bl = b_hid + (((size_t)layer * NN + i) * TT + t) * HM;
        float yreg[2][8];
        for (int ho = 0; ho < 2; ++ho) {
            const int halfbase = ho * 16;
            v8f c;
            #pragma unroll
            for (int j = 0; j < 8; ++j) c[j] = bl[halfbase + hf * 8 + j];
            const float* Arow = Wl + (halfbase + mrow) * HM + koff;
            #pragma unroll
            for (int kc = 0; kc < 8; ++kc) {          // K = 32
                v2f a; a.x = Arow[kc * 4];       a.y = Arow[kc * 4 + 1];
                v2f b; b.x = hin[kc * 4 + koff]; b.y = hin[kc * 4 + koff + 1];
                c = __builtin_amdgcn_wmma_f32_16x16x4_f32(false, a, false, b,
                                                          (short)0, c, false, false);
            }
            #pragma unroll
            for (int j = 0; j < 8; ++j) {
                float v = c[j];
                yreg[ho][j] = (layer == 0 && v < 0.0f) ? 0.0f : v;
            }
        }
        if (mrow == 0) {
            #pragma unroll
            for (int j = 0; j < 8; ++j) {
                hout[hf * 8 + j]      = yreg[0][j];
                hout[16 + hf * 8 + j] = yreg[1][j];
            }
        }
        __builtin_amdgcn_wave_barrier();
        hin  = hout;       // layer0 -> h1, layer1 -> h2
        hout = h2;
    }

    // ---- output: sigmoid(dot(W_out, h) + b_out) ----
    float s = W_out[(size_t)cell * HM + lane] * hin[lane];
    #pragma unroll
    for (int off = 16; off > 0; off >>= 1)
        s += __shfl_xor(s, off, 32);
    if (lane == 0) {
        s += b_out[cell];
        out[cell] = 1.0f / (1.0f + __expf(-s));
    }
}

extern "C" void kernel_launch(void* const* d_in, const int* in_sizes, int n_in,
                              void* d_out, int out_size, void* d_ws, size_t ws_size,
                              hipStream_t stream) {
    const float* X        = (const float*)d_in[0];
    const int*   rows_idx = (const int*)  d_in[1];
    const int*   cols_idx = (const int*)  d_in[2];
    const float* W_in     = (const float*)d_in[3];
    const float* b_in     = (const float*)d_in[4];
    const float* W_hid    = (const float*)d_in[5];
    const float* b_hid    = (const float*)d_in[6];
    const float* W_out    = (const float*)d_in[7];
    const float* b_out    = (const float*)d_in[8];
    float*       out      = (float*)d_out;

    const int cells  = NN * TT;            // 16384
    const int blocks = cells / WPB;        // 2048 blocks x 256 threads (8 wave32/block)
    mar_mlp_kernel<<<blocks, 256, 0, stream>>>(X, rows_idx, cols_idx, W_in, b_in,
                                               W_hid, b_hid, W_out, b_out, out);
    (void)in_sizes; (void)n_in; (void)out_size; (void)d_ws; (void)ws_size;
}